// MemOps_76321568850160
// MI455X (gfx1250) — compile-verified
//
#include <hip/hip_runtime.h>

// NCE memory-bank contrastive kernel for MI455X (gfx1250, wave32).
//
// d_in: [0]=x [16,128] f32, [1]=z [16,128] f32, [2]=y [16] i32,
//       [3]=idx [16,65536] i32, [4]=memory [500000,128] f32
// d_out: lx [16,65537] | lz [16,65537] | new_memory [500000,128], all f32.
//
// Memory-bound (~1.06 GB traffic -> ~45us @ 23.3 TB/s). The batched gathered
// dot-products run as V_WMMA_F32_16X16X4_F32 accumulations; each wave now
// processes TPW tiles so the invariant B matrix (x[b]/z[b] columns) is built
// once per wave and the steady-state loop is pure gather-loads + WMMA.

typedef __attribute__((ext_vector_type(2))) float v2f;
typedef __attribute__((ext_vector_type(4))) float v4f;
typedef __attribute__((ext_vector_type(8))) float v8f;

#define N_DATA 500000
#define N_DIM  128
#define KNEG   65536
#define KP1    65537          // K + 1 logits per batch element
#define BSZ    16
#define NTILES 4097           // ceil(65537 / 16)
#define TPW    8              // tiles per wave
#define NBLKX  513            // ceil(NTILES / TPW)
#define INV_T  (1.0f / 0.07f)

// ---------------------------------------------------------------------------
// Kernel 1: logits via WMMA. One wave32 handles TPW 16-row tiles of gathered
// memory rows for one batch element b:
//   A [16 x 128] = gathered rows (row m -> k-position tile*16+m)
//   B [128 x 16] = col0 = x[b], col1 = z[b], cols 2..15 = 0   (hoisted)
//   D [16 x 16]  accumulated over 32 steps of V_WMMA_F32_16X16X4_F32
//   lx = D[:,0] * (1/T),  lz = D[:,1] * (1/T)
// ---------------------------------------------------------------------------
__global__ __launch_bounds__(32) void nce_logits_wmma(
    const float* __restrict__ x, const float* __restrict__ z,
    const int* __restrict__ y, const int* __restrict__ idx,
    const float* __restrict__ mem,
    float* __restrict__ lx, float* __restrict__ lz)
{
    const int tile0 = blockIdx.x * TPW;     // first tile of this wave
    const int b     = blockIdx.y;           // 0..15
    const int lane  = threadIdx.x;          // 0..31
    const int sub   = lane & 15;            // A: row m; B/D: column N
    const int half  = lane >> 4;            // K/M half selector

    // --- B matrix, invariant across all tiles of this wave ----------------
    // B-matrix 4x16 f32 layout (ISA 7.12.2): VGPR j, lane-half h -> K = 2h+j,
    // N = sub. Step kk covers d = 4*kk..4*kk+3, so this lane holds float2 at
    // d = 4*kk + 2*half. Only columns N=0 (x) and N=1 (z) are non-zero.
    const float* src = (sub == 0) ? (x + (size_t)b * N_DIM)
                                  : (z + (size_t)b * N_DIM);
    v2f bmat[32];
#pragma unroll
    for (int kk = 0; kk < 32; ++kk) {
        v2f t = *(const v2f*)(src + 4 * kk + 2 * half);
        bmat[kk].x = (sub < 2) ? t.x : 0.0f;
        bmat[kk].y = (sub < 2) ? t.y : 0.0f;
    }

    // --- Prefetch gather indices for all TPW tiles -------------------------
    // k-position r==0 is the positive (y[b]); r in [1,65536] are negatives
    // idx[b][r-1]. Out-of-range rows clamp to the positive row (loads stay
    // in-bounds; their stores are masked below).
    const int ypos = y[b];
    int rowidx[TPW];
#pragma unroll
    for (int t = 0; t < TPW; ++t) {
        int r  = (tile0 + t) * 16 + sub;
        int rr = (r < KP1) ? r : 0;
        rowidx[t] = (rr == 0) ? ypos : idx[(size_t)b * KNEG + (rr - 1)];
    }

    // --- Main loop: gather A tile, 32 accumulating WMMAs, store ------------
    for (int t = 0; t < TPW; ++t) {
        int tile = tile0 + t;
        if (tile >= NTILES) break;          // wave-uniform

        // A-matrix 16x4 f32 layout: lanes 0-15 hold K={0,1} in VGPR{0,1},
        // lanes 16-31 hold K={2,3} -> lane loads float2 at d = 4*kk + 2*half.
        const float* aptr = mem + (size_t)rowidx[t] * N_DIM + 2 * half;
        v2f a[32];
#pragma unroll
        for (int kk = 0; kk < 32; ++kk)
            a[kk] = *(const v2f*)(aptr + 4 * kk);

        v8f c = {0.f, 0.f, 0.f, 0.f, 0.f, 0.f, 0.f, 0.f};
#pragma unroll
        for (int kk = 0; kk < 32; ++kk) {
            // 8 args: (neg_a, A, neg_b, B, c_mod, C, reuse_a, reuse_b)
            c = __builtin_amdgcn_wmma_f32_16x16x4_f32(
                    false, a[kk], false, bmat[kk], (short)0, c, false, false);
        }

        // D 16x16 f32 layout: c[v] holds D[8*half + v, N=sub]. Columns 0/1
        // carry the lx/lz dot products for rows m = 0..15 of this tile.
        if (sub < 2) {
            float* outp = (sub == 0) ? lx : lz;
#pragma unroll
            for (int v = 0; v < 8; ++v) {
                int rv = tile * 16 + 8 * half + v;
                if (rv < KP1)
                    outp[(size_t)b * KP1 + rv] = c[v] * INV_T;
            }
        }
    }
}

// ---------------------------------------------------------------------------
// Kernel 2: stream memory -> new_memory (256 MB). Nontemporal float4 copy so
// the 2x256MB stream does not thrash L2.
// ---------------------------------------------------------------------------
__global__ __launch_bounds__(256) void mem_copy_nt(
    const v4f* __restrict__ src, v4f* __restrict__ dst, size_t n4)
{
    size_t i      = (size_t)blockIdx.x * blockDim.x + threadIdx.x;
    size_t stride = (size_t)gridDim.x * blockDim.x;
    for (; i < n4; i += stride) {
        v4f v = __builtin_nontemporal_load(src + i);
        __builtin_nontemporal_store(v, dst + i);
    }
}

// ---------------------------------------------------------------------------
// Kernel 3: overwrite the 16 rows at y with l2norm(0.5*mem[y] + 0.5*x).
// One wave per batch element; 4 floats/lane; shuffle-xor norm reduction.
// Runs after the copy on the same stream.
// ---------------------------------------------------------------------------
__global__ __launch_bounds__(32) void mem_row_update(
    const float* __restrict__ x, const int* __restrict__ y,
    const float* __restrict__ mem, float* __restrict__ out_mem)
{
    const int b    = blockIdx.x;    // 0..15
    const int lane = threadIdx.x;   // 0..31
    const size_t row = (size_t)y[b];

    v4f m4 = *(const v4f*)(mem + row * N_DIM + lane * 4);
    v4f x4 = *(const v4f*)(x + (size_t)b * N_DIM + lane * 4);
    v4f w;
    w.x = m4.x * 0.5f + x4.x * 0.5f;
    w.y = m4.y * 0.5f + x4.y * 0.5f;
    w.z = m4.z * 0.5f + x4.z * 0.5f;
    w.w = m4.w * 0.5f + x4.w * 0.5f;

    float s = w.x * w.x + w.y * w.y + w.z * w.z + w.w * w.w;
#pragma unroll
    for (int off = 16; off > 0; off >>= 1)
        s += __shfl_xor(s, off, 32);

    float inv = 1.0f / fmaxf(sqrtf(s), 1e-12f);
    v4f o;
    o.x = w.x * inv; o.y = w.y * inv; o.z = w.z * inv; o.w = w.w * inv;
    *(v4f*)(out_mem + row * N_DIM + lane * 4) = o;
}

// ---------------------------------------------------------------------------
extern "C" void kernel_launch(void* const* d_in, const int* in_sizes, int n_in,
                              void* d_out, int out_size, void* d_ws, size_t ws_size,
                              hipStream_t stream)
{
    const float* x   = (const float*)d_in[0];
    const float* z   = (const float*)d_in[1];
    const int*   y   = (const int*)d_in[2];
    const int*   idx = (const int*)d_in[3];
    const float* mem = (const float*)d_in[4];

    float* lx = (float*)d_out;
    float* lz = lx + (size_t)BSZ * KP1;
    float* nm = lz + (size_t)BSZ * KP1;

    // Logits: 513 x 16 wave32 blocks, TPW tiles per wave.
    dim3 g1(NBLKX, BSZ);
    nce_logits_wmma<<<g1, dim3(32), 0, stream>>>(x, z, y, idx, mem, lx, lz);

    // new_memory = copy of memory (nontemporal stream), then EMA row updates.
    size_t n4 = (size_t)N_DATA * N_DIM / 4;   // 16M float4
    mem_copy_nt<<<dim3(8192), dim3(256), 0, stream>>>((const v4f*)mem, (v4f*)nm, n4);
    mem_row_update<<<dim3(BSZ), dim3(32), 0, stream>>>(x, y, mem, nm);
}